// ClimbGenerator_592705487394
// MI455X (gfx1250) — compile-verified
//
#include <hip/hip_runtime.h>

typedef __attribute__((ext_vector_type(2))) float v2f;
typedef __attribute__((ext_vector_type(8))) float v8f;

// ---------------------------------------------------------------- zero
__global__ void gc_zero_kernel(float* __restrict__ p, int count) {
    int i = blockIdx.x * blockDim.x + threadIdx.x;
    if (i < count) p[i] = 0.0f;
}

// ---------------------------------------------------------------- scatter-add
// One thread per edge: agg[dst] += w * x[src]  (din-wide).
// agg is L2-resident (<= 8 MB); global_atomic_add_f32 RMW happens in L2.
template<int DIN>
__global__ __launch_bounds__(256)
void gc_scatter_kernel(const float* __restrict__ x,
                       const int* __restrict__ src,
                       const int* __restrict__ dst,
                       const float* __restrict__ ew,
                       float* __restrict__ agg,
                       int E) {
    int e = blockIdx.x * blockDim.x + threadIdx.x;
    if (e >= E) return;
    int s = src[e];
    int d = dst[e];
    float w = ew[e];
    const float* xs = x + (long)s * DIN;
    float* ad = agg + (long)d * DIN;
#pragma unroll
    for (int k = 0; k < DIN; ++k) {
        atomicAdd(&ad[k], w * xs[k]);
    }
}

// ---------------------------------------------------------------- dense (WMMA)
// y[16 rows x DOUT] = [agg | x] (16 x 2*DIN) @ [W_rel ; W_root] (2*DIN x DOUT) + b
// One wave32 per 16-row node tile, V_WMMA_F32_16X16X4_F32 over K chunks of 4.
template<int DIN, int DOUT, bool RELU>
__global__ __launch_bounds__(256)
void gc_dense_kernel(const float* __restrict__ agg,
                     const float* __restrict__ x,
                     const float* __restrict__ Wrel,   // [DIN][DOUT] row-major
                     const float* __restrict__ brel,   // [DOUT]
                     const float* __restrict__ Wroot,  // [DIN][DOUT] row-major
                     float* __restrict__ y,            // [n][DOUT]
                     int n) {
    constexpr int K   = 2 * DIN;
    constexpr int KP  = (K + 3) & ~3;   // K padded to multiple of 4
    constexpr int NKC = KP / 4;         // number of K chunks
    constexpr int CT  = (DOUT + 15) / 16;

    const int lane = threadIdx.x & 31;
    const int wave = blockIdx.x * (blockDim.x >> 5) + (threadIdx.x >> 5);
    const int row0 = wave * 16;
    if (row0 >= n) return;              // wave-uniform exit (n % 16 == 0)

    const int half = lane >> 4;         // K half selector
    const int lr   = lane & 15;         // A: row-in-tile, B/C: column
    const long arow = row0 + lr;

    // ---- A fragments: rows of [agg | x], layout per 32-bit 16x4 A matrix
    v2f afrag[NKC];
#pragma unroll
    for (int kc = 0; kc < NKC; ++kc) {
#pragma unroll
        for (int j = 0; j < 2; ++j) {
            const int k = kc * 4 + half * 2 + j;   // k depends on lane half
            float av = 0.0f;
            if (k < DIN)            av = agg[arow * DIN + k];
            else if (k < 2 * DIN)   av = x[arow * DIN + (k - DIN)];
            afrag[kc][j] = av;
        }
    }

#pragma unroll
    for (int ct = 0; ct < CT; ++ct) {
        const int col = ct * 16 + lr;
        v8f c = {};
#pragma unroll
        for (int kc = 0; kc < NKC; ++kc) {
            // ---- B fragment: [W_rel ; W_root] column `col`, same K split
            v2f bfrag;
#pragma unroll
            for (int j = 0; j < 2; ++j) {
                const int k = kc * 4 + half * 2 + j;
                float bv = 0.0f;
                if (col < DOUT) {
                    if (k < DIN)          bv = Wrel[k * DOUT + col];
                    else if (k < 2 * DIN) bv = Wroot[(k - DIN) * DOUT + col];
                }
                bfrag[j] = bv;
            }
            // EXEC is all-1s here (guards rejoin before the WMMA)
            c = __builtin_amdgcn_wmma_f32_16x16x4_f32(
                    /*neg_a=*/false, afrag[kc],
                    /*neg_b=*/false, bfrag,
                    /*c_mod=*/(short)0, c,
                    /*reuse_a=*/false, /*reuse_b=*/false);
        }
        // ---- epilogue: bias + ReLU, store
        if (col < DOUT) {
            const float bias = brel[col];
#pragma unroll
            for (int r = 0; r < 8; ++r) {
                const long row = row0 + r + 8 * half;
                float val = c[r] + bias;
                if (RELU) val = fmaxf(val, 0.0f);
                y[row * DOUT + col] = val;
            }
        }
    }
}

// ---------------------------------------------------------------- softmax (dout=2)
__global__ void gc_softmax2_kernel(float* __restrict__ y, int n) {
    int i = blockIdx.x * blockDim.x + threadIdx.x;
    if (i >= n) return;
    float a = y[2 * i + 0];
    float b = y[2 * i + 1];
    float m = fmaxf(a, b);
    float ea = __expf(a - m);
    float eb = __expf(b - m);
    float s = ea + eb;
    y[2 * i + 0] = ea / s;
    y[2 * i + 1] = eb / s;
}

// ---------------------------------------------------------------- host helpers
template<int DIN, int DOUT, bool RELU>
static void gc_run_layer(const float* xin, const int* src, const int* dst,
                         const float* ew, int E, int n,
                         const float* wrel, const float* brel, const float* wroot,
                         float* agg, float* yout, hipStream_t stream) {
    const int zc = n * DIN;
    gc_zero_kernel<<<(zc + 255) / 256, 256, 0, stream>>>(agg, zc);
    gc_scatter_kernel<DIN><<<(E + 255) / 256, 256, 0, stream>>>(xin, src, dst, ew, agg, E);
    const int waves  = (n + 15) / 16;
    const int blocks = (waves + 7) / 8;   // 8 waves (256 threads) per block
    gc_dense_kernel<DIN, DOUT, RELU><<<blocks, 256, 0, stream>>>(
        agg, xin, wrel, brel, wroot, yout, n);
}

extern "C" void kernel_launch(void* const* d_in, const int* in_sizes, int n_in,
                              void* d_out, int out_size, void* d_ws, size_t ws_size,
                              hipStream_t stream) {
    const float* x0 = (const float*)d_in[0];
    const int*   ei = (const int*)d_in[1];
    const float* ew = (const float*)d_in[2];
    const int E = in_sizes[2];
    const int n = in_sizes[0] / 6;
    const int* src = ei;
    const int* dst = ei + E;

    // workspace: agg[n*20] | actA[n*20] | actB[n*20]  (24 MB total)
    float* agg  = (float*)d_ws;
    float* actA = agg  + (size_t)n * 20;
    float* actB = actA + (size_t)n * 20;
    float* out  = (float*)d_out;

    gc_run_layer<6, 20, true >(x0,   src, dst, ew, E, n,
        (const float*)d_in[3],  (const float*)d_in[4],  (const float*)d_in[5],
        agg, actA, stream);
    gc_run_layer<20, 15, true>(actA, src, dst, ew, E, n,
        (const float*)d_in[6],  (const float*)d_in[7],  (const float*)d_in[8],
        agg, actB, stream);
    gc_run_layer<15, 10, true>(actB, src, dst, ew, E, n,
        (const float*)d_in[9],  (const float*)d_in[10], (const float*)d_in[11],
        agg, actA, stream);
    gc_run_layer<10, 5, true >(actA, src, dst, ew, E, n,
        (const float*)d_in[12], (const float*)d_in[13], (const float*)d_in[14],
        agg, actB, stream);
    gc_run_layer<5, 2, false>(actB, src, dst, ew, E, n,
        (const float*)d_in[15], (const float*)d_in[16], (const float*)d_in[17],
        agg, out, stream);

    gc_softmax2_kernel<<<(n + 255) / 256, 256, 0, stream>>>(out, n);
}